// RoutingBlock_12575664243335
// MI455X (gfx1250) — compile-verified
//
#include <hip/hip_runtime.h>
#include <math.h>

#define Bsz 2048
#define Dsz 512
#define Nsz 128
#define Esz 16
#define Ksz 4

typedef float v2f __attribute__((ext_vector_type(2)));
typedef float v4f __attribute__((ext_vector_type(4)));
typedef float v8f __attribute__((ext_vector_type(8)));

// ---------------------------------------------------------------------------
// Kernel 0: zero the 16-float "load" accumulator region of d_out.
// ---------------------------------------------------------------------------
__global__ void k_zero_load(float* __restrict__ loadv) {
  if (threadIdx.x < Esz) loadv[threadIdx.x] = 0.0f;
}

// ---------------------------------------------------------------------------
// Kernel 1 (bandwidth-bound, ~512 MB streamed): x[b,d] = <x_trans[b,d,:], w> + b0
// One wave32 per (b,d) row: lane i loads floats [4i,4i+4) (coalesced b128,
// non-temporal), 4-FMA partial dot, 5-step shfl_xor butterfly reduction.
// ---------------------------------------------------------------------------
__global__ void __launch_bounds__(256) k_start_linear(
    const float* __restrict__ xt, const float* __restrict__ sw,
    const float* __restrict__ sb, float* __restrict__ x) {
  const int lane = threadIdx.x & 31;
  const int wave = (int)((blockIdx.x * blockDim.x + threadIdx.x) >> 5);
  const int nw   = (int)((gridDim.x * blockDim.x) >> 5);
  const v4f w    = ((const v4f*)sw)[lane];   // 128 floats = 32 x float4
  const float b0 = sb[0];
  const int rows = Bsz * Dsz;
  for (int row = wave; row < rows; row += nw) {
    const v4f* p = (const v4f*)(xt + (size_t)row * Nsz);
    v4f a = __builtin_nontemporal_load(p + lane);
    float s = a.x * w.x + a.y * w.y + a.z * w.z + a.w * w.w;
#pragma unroll
    for (int off = 16; off > 0; off >>= 1) s += __shfl_xor(s, off, 32);
    if (lane == 0) x[row] = s + b0;
  }
}

__device__ __forceinline__ float softplusf(float z) {
  return (z > 20.0f) ? z : log1pf(expf(z));
}

// ---------------------------------------------------------------------------
// Kernel 2 (WMMA): clean/noise logits via V_WMMA_F32_16X16X4_F32, then per-row
// noisy top-5, softmax gates, and Normal-CDF load accumulation.
// One wave (32 threads) per 16-row tile of B.
// ---------------------------------------------------------------------------
__global__ void __launch_bounds__(32) k_gate(
    const float* __restrict__ x, const float* __restrict__ noise,
    const float* __restrict__ gate_w, const float* __restrict__ gate_b,
    const float* __restrict__ noise_w, const float* __restrict__ noise_b,
    float* __restrict__ gates, float* __restrict__ loadv) {
  __shared__ float clean_s[16][17];
  __shared__ float std_s[16][17];
  __shared__ float load_s[16][16];

  const int lane = (int)threadIdx.x;  // 0..31
  const int half = lane >> 4;         // 0 or 1
  const int l    = lane & 15;
  const int b0   = (int)blockIdx.x * 16;

  v8f cg = {0.f, 0.f, 0.f, 0.f, 0.f, 0.f, 0.f, 0.f};
  v8f cn = {0.f, 0.f, 0.f, 0.f, 0.f, 0.f, 0.f, 0.f};

  // A fragment: lane -> row M=l of the tile, K pair {2*half, 2*half+1}
  const float* arow = x       + (size_t)(b0 + l) * Dsz + 2 * half;
  // B fragment: lane -> column N=l (expert), same K pair; B[k][n] = w[n][d0+k]
  const float* grow = gate_w  + (size_t)l * Dsz + 2 * half;
  const float* nrow = noise_w + (size_t)l * Dsz + 2 * half;

#pragma unroll 4
  for (int d0 = 0; d0 < Dsz; d0 += 4) {
    v2f a  = *(const v2f*)(arow + d0);
    v2f bg = *(const v2f*)(grow + d0);
    v2f bn = *(const v2f*)(nrow + d0);
    cg = __builtin_amdgcn_wmma_f32_16x16x4_f32(false, a, false, bg, (short)0,
                                               cg, false, false);
    cn = __builtin_amdgcn_wmma_f32_16x16x4_f32(false, a, false, bn, (short)0,
                                               cn, false, false);
  }

  // C layout: VGPR v holds row M = v + 8*half, column N = l.
  const float gb = gate_b[l];
  const float nb = noise_b[l];
#pragma unroll
  for (int v = 0; v < 8; ++v) {
    const int m   = v + 8 * half;
    clean_s[m][l] = cg[v] + gb;
    std_s[m][l]   = softplusf(cn[v] + nb) + 0.01f;
  }
  __syncthreads();

  if (lane < 16) {
    const int r = lane;
    const int b = b0 + r;
    float cl[16], sd[16], nl[16];
#pragma unroll
    for (int e = 0; e < 16; ++e) {
      cl[e] = clean_s[r][e];
      sd[e] = std_s[r][e];
      nl[e] = cl[e] + noise[(size_t)b * Esz + e] * sd[e];
    }
    // top-5 selection over 16 noisy logits
    float tv[5]; int ti[5];
    unsigned used = 0u;
#pragma unroll
    for (int s = 0; s < 5; ++s) {
      float m = -INFINITY; int mi = 0;
      for (int e = 0; e < 16; ++e)
        if (!(used & (1u << e)) && nl[e] > m) { m = nl[e]; mi = e; }
      used |= (1u << mi);
      tv[s] = m; ti[s] = mi;
    }
    // softmax over top-4 (tv[0] is the max)
    float ex[4], ssum = 0.0f;
#pragma unroll
    for (int s = 0; s < 4; ++s) { ex[s] = expf(tv[s] - tv[0]); ssum += ex[s]; }
    const float inv = 1.0f / ssum;
    float g[16];
#pragma unroll
    for (int e = 0; e < 16; ++e) g[e] = 0.0f;
#pragma unroll
    for (int s = 0; s < 4; ++s) g[ti[s]] = ex[s] * inv;
    v4f* gout = (v4f*)(gates + (size_t)b * Esz);
#pragma unroll
    for (int q = 0; q < 4; ++q) {
      v4f t; t.x = g[4*q]; t.y = g[4*q+1]; t.z = g[4*q+2]; t.w = g[4*q+3];
      gout[q] = t;
    }
    // load: Phi((clean - thr) / sigma), thr depends on in/out of top-K
    const float thr_in = tv[4], thr_out = tv[3];
    const float inv_sqrt2 = 0.70710678118654752f;
#pragma unroll
    for (int e = 0; e < 16; ++e) {
      const float thr = (nl[e] > thr_in) ? thr_in : thr_out;
      const float z   = (cl[e] - thr) / sd[e];
      load_s[r][e]    = 0.5f * (1.0f + erff(z * inv_sqrt2));
    }
  }
  __syncthreads();
  if (lane < 16) {
    float s = 0.0f;
#pragma unroll
    for (int r = 0; r < 16; ++r) s += load_s[r][lane];
    atomicAdd(&loadv[lane], s);
  }
}

// ---------------------------------------------------------------------------
extern "C" void kernel_launch(void* const* d_in, const int* in_sizes, int n_in,
                              void* d_out, int out_size, void* d_ws,
                              size_t ws_size, hipStream_t stream) {
  const float* xt = (const float*)d_in[0];  // [B, D, N]
  const float* nz = (const float*)d_in[1];  // [B, E]
  const float* sw = (const float*)d_in[2];  // [1, N]
  const float* sb = (const float*)d_in[3];  // [1]
  const float* gw = (const float*)d_in[4];  // [E, D]
  const float* gb = (const float*)d_in[5];  // [E]
  const float* nw = (const float*)d_in[6];  // [E, D]
  const float* nb = (const float*)d_in[7];  // [E]

  float* gates = (float*)d_out;             // [B, E]
  float* loadv = (float*)d_out + Bsz * Esz; // [E]
  float* x     = (float*)d_ws;              // [B, D] scratch (4 MB)

  hipLaunchKernelGGL(k_zero_load, dim3(1), dim3(32), 0, stream, loadv);
  hipLaunchKernelGGL(k_start_linear, dim3(4096), dim3(256), 0, stream,
                     xt, sw, sb, x);
  hipLaunchKernelGGL(k_gate, dim3(Bsz / 16), dim3(32), 0, stream,
                     x, nz, gw, gb, nw, nb, gates, loadv);
}